// P2MLoss_2465311228497
// MI455X (gfx1250) — compile-verified
//
#include <hip/hip_runtime.h>
#include <stdint.h>

// ---------------------------------------------------------------------------
// P2M loss for MI455X (gfx1250). Chamfer pairwise-distance core runs on
// V_WMMA_F32_16X16X4_F32 (16x16 tile, K=4, FP32). Branchless min/argmin
// epilogue (v_cndmask only, no EXEC manipulation around WMMA), float4-packed
// pred points (one global_load_b128 per B tile), software-pipelined loads.
// ---------------------------------------------------------------------------

#define NGT   8000
#define NSAMP 4000
#define RT    10                 // 10 row tiles of 16 -> 160 GT rows per wave
#define ROWS_PER_WG (RT * 16)
#define COL_TILES_PER_WG 32      // 512 pred columns per workgroup chunk

typedef float v2f __attribute__((ext_vector_type(2)));
typedef float v8f __attribute__((ext_vector_type(8)));

// ---- monotonic float <-> uint key (ascending order preserved) -------------
__device__ __forceinline__ unsigned fkey(float f) {
  unsigned u = __float_as_uint(f);
  return (u & 0x80000000u) ? ~u : (u | 0x80000000u);
}
__device__ __forceinline__ float funkey(unsigned k) {
  unsigned u = (k & 0x80000000u) ? (k & 0x7fffffffu) : ~k;
  return __uint_as_float(u);
}

// ---- cheap counter-based PRNG (JAX threefry is not reproducible anyway) ---
__device__ __forceinline__ unsigned pcg(unsigned x) {
  x = x * 747796405u + 2891336453u;
  unsigned w = ((x >> ((x >> 28) + 4u)) ^ x) * 277803737u;
  return (w >> 22) ^ w;
}
__device__ __forceinline__ float u01(unsigned h) {
  return (float)(h >> 8) * (1.0f / 16777216.0f);
}

// ---------------------------------------------------------------------------
__global__ void k_init_out(float* out) {
  if (threadIdx.x == 0 && blockIdx.x == 0) out[0] = 0.0f;
}

__global__ void k_gtnorm(const float* __restrict__ gt, float* __restrict__ gtn2) {
  int n = blockIdx.x * blockDim.x + threadIdx.x;
  if (n < NGT) {
    float x = gt[3 * n], y = gt[3 * n + 1], z = gt[3 * n + 2];
    gtn2[n] = x * x + y * y + z * z;
  }
}

__global__ void k_init_level(unsigned* __restrict__ d1k,
                             unsigned long long* __restrict__ d2k,
                             float* __restrict__ accum, int Mpad) {
  int i = blockIdx.x * blockDim.x + threadIdx.x;
  if (i < NGT)  d1k[i] = fkey(INFINITY);
  if (i < Mpad) d2k[i] = 0xFFFFFFFFFFFFFFFFull;
  if (i < 8)    accum[i] = 0.0f;
}

// ---- sample_pred packed as float4 (x, y, z, |p|^2) ------------------------
__global__ void k_pack(const float* __restrict__ pred, const int* __restrict__ faces,
                       int V, int F, int M, int level, float4* __restrict__ P4) {
  int m = blockIdx.x * blockDim.x + threadIdx.x;
  if (m >= M) return;
  float x, y, z;
  if (m < V) {
    x = pred[3 * m]; y = pred[3 * m + 1]; z = pred[3 * m + 2];
  } else {
    unsigned s = (unsigned)(level * NSAMP + (m - V));
    int f = (int)(pcg(s * 3u + 1u) % (unsigned)F);
    float u1 = u01(pcg(s * 3u + 2u));
    float u2 = u01(pcg(s * 3u + 3u));
    float r1 = sqrtf(u1);
    float w0 = 1.0f - r1, w1 = r1 * (1.0f - u2), w2 = r1 * u2;
    int i0 = faces[3 * f], i1 = faces[3 * f + 1], i2 = faces[3 * f + 2];
    x = w0 * pred[3 * i0] + w1 * pred[3 * i1] + w2 * pred[3 * i2];
    y = w0 * pred[3 * i0 + 1] + w1 * pred[3 * i1 + 1] + w2 * pred[3 * i2 + 1];
    z = w0 * pred[3 * i0 + 2] + w1 * pred[3 * i1 + 2] + w2 * pred[3 * i2 + 2];
  }
  P4[m] = make_float4(x, y, z, x * x + y * y + z * z);
}

// ---------------------------------------------------------------------------
// Chamfer core. One wave per (160-row strip, 512-col chunk).
// A tiles (16x4 f32, 2 VGPR): a[j][lane] = gt[rowBase+lane%16][k= j + 2*(lane/16)]
// B tile  (4x16 f32, 2 VGPR): b[j][lane] = P [colBase+lane%16][k= j + 2*(lane/16)]
// D (16x16 f32, 8 VGPR):      D[r][lane] = dot(row r+8*(lane/16), col lane%16)
// ---------------------------------------------------------------------------
__global__ __launch_bounds__(32)
void k_chamfer(const float* __restrict__ gt, const float* __restrict__ gtn2,
               const float4* __restrict__ P4,
               unsigned* __restrict__ d1k, unsigned long long* __restrict__ d2k,
               int M, int Mtiles) {
  const int lane = threadIdx.x;
  const int hi   = lane >> 4;        // 0 = rows r, 1 = rows r+8
  const int hi8  = hi << 3;
  const int ml   = lane & 15;
  const int rowBase = blockIdx.y * ROWS_PER_WG;   // NGT = 50 * 160 exactly
  const int ridBase = rowBase + hi8;

  v2f   A[RT];
  float gnr[RT][8];                  // |gt|^2 broadcast per (tile,reg,half)
  float rmin[RT][8];                 // running min of (|p|^2 - 2 g.p) per row
#pragma unroll
  for (int rt = 0; rt < RT; ++rt) {
    int row = rowBase + rt * 16 + ml;
    float gy = gt[3 * row + 1];                     // always-valid load
    A[rt].x = gt[3 * row + 2 * hi];                 // k = 0 or 2
    A[rt].y = hi ? 0.0f : gy;                       // k = 1 or 3 (pad)
    float gnl = gtn2[row];
#pragma unroll
    for (int r = 0; r < 8; ++r) {
      gnr[rt][r]  = __shfl(gnl, r + hi8, 32);
      rmin[rt][r] = INFINITY;
    }
  }

  const int ctBegin = blockIdx.x * COL_TILES_PER_WG;
  const int ctEnd   = min(ctBegin + COL_TILES_PER_WG, Mtiles);

  // software pipeline: q holds this tile's point, fetch next tile early
  float4 q = P4[min(ctBegin * 16 + ml, M - 1)];

  for (int ct = ctBegin; ct < ctEnd; ++ct) {        // wave-uniform loop
    const int nextCol = min(min(ct + 1, ctEnd - 1) * 16 + ml, M - 1);
    const float4 qn = P4[nextCol];                  // prefetch next B tile

    v2f B;
    B.x = hi ? q.z : q.x;                           // k = 0 or 2 (cndmask)
    B.y = hi ? 0.0f : q.y;                          // k = 1 or 3 (pad)
    const float pnl = q.w;

    float cmin = INFINITY;                          // min over rows of |g|^2-2g.p
    int   cidx = 0;                                 // local row code rt*16+r
#pragma unroll
    for (int rt = 0; rt < RT; ++rt) {
      v8f C = {0.f, 0.f, 0.f, 0.f, 0.f, 0.f, 0.f, 0.f};
      v8f D = __builtin_amdgcn_wmma_f32_16x16x4_f32(
          false, A[rt], false, B, (short)0, C, false, false);
#pragma unroll
      for (int r = 0; r < 8; ++r) {
        float d = D[r];
        rmin[rt][r] = fminf(rmin[rt][r], fmaf(-2.0f, d, pnl));  // dist1 path
        float u = fmaf(-2.0f, d, gnr[rt][r]);                   // dist2 path
        bool lt = u < cmin;                         // strict: keeps first hit
        cidx = lt ? (rt * 16 + r) : cidx;           // cndmask w/ inline const
        cmin = fminf(u, cmin);
      }
    }
    // globalize index, fold the two 16-lane halves (rows m and m+8)
    int   cidxG = cidx + ridBase;
    float ov = __shfl_xor(cmin, 16, 32);
    int   oi = __shfl_xor(cidxG, 16, 32);
    int take = (int)(ov < cmin) | ((int)(ov == cmin) & (int)(oi < cidxG));
    cmin  = take ? ov : cmin;
    cidxG = take ? oi : cidxG;
    int c = ct * 16 + ml;
    if (hi == 0 && c < M) {
      unsigned long long key =
          ((unsigned long long)fkey(cmin + pnl) << 32) | (unsigned)cidxG;
      atomicMin(&d2k[c], key);
    }
    q = qn;
  }

  // dist1: reduce each running-min across the 16 columns held by this half
#pragma unroll
  for (int rt = 0; rt < RT; ++rt) {
#pragma unroll
    for (int r = 0; r < 8; ++r) {
      float v = rmin[rt][r];
      v = fminf(v, __shfl_xor(v, 1, 32));
      v = fminf(v, __shfl_xor(v, 2, 32));
      v = fminf(v, __shfl_xor(v, 4, 32));
      v = fminf(v, __shfl_xor(v, 8, 32));
      if (ml == 0) {                                // lanes 0 and 16 write
        int row = rowBase + rt * 16 + r + hi8;
        atomicMin(&d1k[row], fkey(v + gtn2[row]));
      }
    }
  }
}

// ---- sum dist1/dist2 ------------------------------------------------------
__global__ void k_reduce(const unsigned* __restrict__ d1k,
                         const unsigned long long* __restrict__ d2k,
                         int M, float* __restrict__ accum) {
  int i = blockIdx.x * blockDim.x + threadIdx.x;
  float v1 = 0.0f, v2 = 0.0f;
  if (i < NGT) v1 = funkey(d1k[i]);
  if (i < M)   v2 = funkey((unsigned)(d2k[i] >> 32));
  for (int o = 16; o > 0; o >>= 1) {
    v1 += __shfl_xor(v1, o, 32);
    v2 += __shfl_xor(v2, o, 32);
  }
  if ((threadIdx.x & 31) == 0) {
    atomicAdd(&accum[0], v1);
    atomicAdd(&accum[1], v2);
  }
}

// ---- edge length + normal consistency (uses idx2 packed in d2k) -----------
__global__ void k_edge_normal(const float* __restrict__ pred, const int* __restrict__ edges,
                              int E, const float* __restrict__ gtn,
                              const unsigned long long* __restrict__ d2k,
                              float* __restrict__ accum) {
  int j = blockIdx.x * blockDim.x + threadIdx.x;
  float es = 0.0f, ns = 0.0f;
  if (j < E) {
    int e0 = edges[2 * j], e1 = edges[2 * j + 1];
    float dx = pred[3 * e0] - pred[3 * e1];
    float dy = pred[3 * e0 + 1] - pred[3 * e1 + 1];
    float dz = pred[3 * e0 + 2] - pred[3 * e1 + 2];
    float q = dx * dx + dy * dy + dz * dz;
    es = q;
    float inv = 1.0f / fmaxf(sqrtf(q), 1e-12f);
    unsigned i2 = (unsigned)(d2k[e0] & 0xffffffffu);
    float nx = gtn[3 * i2], ny = gtn[3 * i2 + 1], nz = gtn[3 * i2 + 2];
    float nn = 1.0f / fmaxf(sqrtf(nx * nx + ny * ny + nz * nz), 1e-12f);
    ns = fabsf((dx * nx + dy * ny + dz * nz) * inv * nn);
  }
  for (int o = 16; o > 0; o >>= 1) {
    es += __shfl_xor(es, o, 32);
    ns += __shfl_xor(ns, o, 32);
  }
  if ((threadIdx.x & 31) == 0) {
    atomicAdd(&accum[2], ns);
    atomicAdd(&accum[3], es);
  }
}

// ---- laplacian + move terms ----------------------------------------------
__global__ void k_lap_move(const float* __restrict__ pred, const float* __restrict__ bef,
                           const int* __restrict__ lap, int V, float* __restrict__ accum) {
  int v = blockIdx.x * blockDim.x + threadIdx.x;
  float ls = 0.0f, ms = 0.0f;
  if (v < V) {
    float sx = 0, sy = 0, sz = 0, bx = 0, by = 0, bz = 0;
#pragma unroll
    for (int k = 0; k < 8; ++k) {
      int idx = lap[10 * v + k];
      int s = idx < 0 ? 0 : idx;
      float w = idx < 0 ? 0.0f : 1.0f;
      sx += w * pred[3 * s]; sy += w * pred[3 * s + 1]; sz += w * pred[3 * s + 2];
      bx += w * bef[3 * s];  by += w * bef[3 * s + 1];  bz += w * bef[3 * s + 2];
    }
    float inv = 1.0f / (float)lap[10 * v + 9];
    float lpx = pred[3 * v] - sx * inv, lpy = pred[3 * v + 1] - sy * inv,
          lpz = pred[3 * v + 2] - sz * inv;
    float lbx = bef[3 * v] - bx * inv,  lby = bef[3 * v + 1] - by * inv,
          lbz = bef[3 * v + 2] - bz * inv;
    float dx = lbx - lpx, dy = lby - lpy, dz = lbz - lpz;
    ls = dx * dx + dy * dy + dz * dz;
    float mx = bef[3 * v] - pred[3 * v], my = bef[3 * v + 1] - pred[3 * v + 1],
          mz = bef[3 * v + 2] - pred[3 * v + 2];
    ms = mx * mx + my * my + mz * mz;
  }
  for (int o = 16; o > 0; o >>= 1) {
    ls += __shfl_xor(ls, o, 32);
    ms += __shfl_xor(ms, o, 32);
  }
  if ((threadIdx.x & 31) == 0) {
    atomicAdd(&accum[4], ls);
    atomicAdd(&accum[5], ms);
  }
}

// ---- per-level weighted accumulation (stream-ordered, single thread) ------
__global__ void k_finalize(const float* __restrict__ accum, float* __restrict__ out,
                           float cham_w, float lap_c, int M, int E, int V, int level) {
  if (threadIdx.x != 0 || blockIdx.x != 0) return;
  float ch = cham_w * (accum[0] / (float)NGT + 0.55f * accum[1] / (float)M);
  float nl = accum[2] / (float)E;
  float el = accum[3] / (float)E;          // mse * 3 == sum/E
  float ll = lap_c * accum[4] / (float)V;  // lap_c * mse * 3
  float ml = (level > 0) ? lap_c * accum[5] / (float)V : 0.0f;
  out[0] += ch + 1500.0f * ll + 50.0f * ml + 359.0f * el + 0.5f * nl;
}

// ---------------------------------------------------------------------------
extern "C" void kernel_launch(void* const* d_in, const int* in_sizes, int n_in,
                              void* d_out, int out_size, void* d_ws, size_t ws_size,
                              hipStream_t stream) {
  const float* gt  = (const float*)d_in[0];
  const float* gtn = (const float*)d_in[1];

  // workspace layout (max M = 10242+4000 = 14242, padded to 14256)
  char* ws = (char*)d_ws;
  float4*             P4   = (float4*)(ws + 0);       // 14256 * 16B = 228096
  float*              gtn2 = (float*)(ws + 229376);   // 8000 f32
  unsigned*           d1k  = (unsigned*)(ws + 262144);            // 8000 u32
  unsigned long long* d2k  = (unsigned long long*)(ws + 294912);  // 14256 u64
  float*              acc  = (float*)(ws + 409600);   // 8 f32
  float* out = (float*)d_out;

  k_init_out<<<1, 1, 0, stream>>>(out);
  k_gtnorm<<<(NGT + 255) / 256, 256, 0, stream>>>(gt, gtn2);

  const float cham_w[3] = {3000.0f, 3000.0f, 3000.0f};
  const float lap_c[3]  = {0.2f, 1.0f, 1.0f};

  for (int i = 0; i < 3; ++i) {
    const float* pred  = (const float*)d_in[2 + 5 * i];
    const float* bef   = (const float*)d_in[3 + 5 * i];
    const int*   edges = (const int*)d_in[4 + 5 * i];
    const int*   faces = (const int*)d_in[5 + 5 * i];
    const int*   lap   = (const int*)d_in[6 + 5 * i];
    int V = in_sizes[2 + 5 * i] / 3;
    int E = in_sizes[4 + 5 * i] / 2;
    int F = in_sizes[5 + 5 * i] / 3;
    int M = V + NSAMP;
    int Mtiles = (M + 15) / 16;
    int Mpad = Mtiles * 16;
    int ninit = Mpad > NGT ? Mpad : NGT;

    k_init_level<<<(ninit + 255) / 256, 256, 0, stream>>>(d1k, d2k, acc, Mpad);
    k_pack<<<(M + 255) / 256, 256, 0, stream>>>(pred, faces, V, F, M, i, P4);

    dim3 grid((Mtiles + COL_TILES_PER_WG - 1) / COL_TILES_PER_WG, NGT / ROWS_PER_WG);
    k_chamfer<<<grid, 32, 0, stream>>>(gt, gtn2, P4, d1k, d2k, M, Mtiles);

    int nred = M > NGT ? M : NGT;
    k_reduce<<<(nred + 255) / 256, 256, 0, stream>>>(d1k, d2k, M, acc);
    k_edge_normal<<<(E + 255) / 256, 256, 0, stream>>>(pred, edges, E, gtn, d2k, acc);
    k_lap_move<<<(V + 255) / 256, 256, 0, stream>>>(pred, bef, lap, V, acc);
    k_finalize<<<1, 1, 0, stream>>>(acc, out, cham_w[i], lap_c[i], M, E, V, i);
  }
}